// SelectiveSSM_40020505264686
// MI455X (gfx1250) — compile-verified
//
#include <hip/hip_runtime.h>
#include <hip/hip_bf16.h>

typedef __attribute__((ext_vector_type(2))) float v2f;
typedef __attribute__((ext_vector_type(8))) float v8f;

// ---------------------------------------------------------------------------
// Fp32 WMMA GEMM:  C[M,N] = A[M,K] @ B[N,K]^T   (row-major fp32)
// Block = 256 threads = 8 waves arranged 4(M) x 2(N); each wave computes a
// 32x64 tile as 2x4 V_WMMA_F32_16X16X4_F32 tiles (64 acc VGPRs).
// Per K-step(4): 6 fragment loads (2 A + 4 B) feed 8 WMMAs.
// Fragment layout per CDNA5 ISA:
//   A 16x4 f32: lanes 0-15 -> M=lane, VGPR0=K0,VGPR1=K1; lanes 16-31 -> K2,K3
//   B symmetric with N in place of M; C: VGPR v -> M=v (lanes<16) / v+8, N=lane&15
// ---------------------------------------------------------------------------
__global__ __launch_bounds__(256) void wmma_gemm_f32(
    const float* __restrict__ A, const float* __restrict__ B,
    float* __restrict__ C, int M, int N, int K) {
  const int lane  = threadIdx.x & 31;
  const int wave  = threadIdx.x >> 5;   // 0..7
  const int waveM = wave >> 1;          // 0..3
  const int waveN = wave & 1;           // 0..1
  const int r     = lane & 15;          // M (for A) / N (for B) index
  const int kh    = lane >> 4;          // selects K pair {0,1} vs {2,3}

  const int mBase = blockIdx.x * 128 + waveM * 32;
  const int nBase = blockIdx.y * 128 + waveN * 64;

  v8f acc[2][4];
  #pragma unroll
  for (int i = 0; i < 2; ++i)
    #pragma unroll
    for (int j = 0; j < 4; ++j) acc[i][j] = (v8f){};

  const float* Ar0 = A + (size_t)(mBase + r)      * K;
  const float* Ar1 = A + (size_t)(mBase + 16 + r) * K;
  const float* Br0 = B + (size_t)(nBase +      r) * K;
  const float* Br1 = B + (size_t)(nBase + 16 + r) * K;
  const float* Br2 = B + (size_t)(nBase + 32 + r) * K;
  const float* Br3 = B + (size_t)(nBase + 48 + r) * K;

  #pragma unroll 2
  for (int kk = 0; kk < K; kk += 4) {
    const int ko = kk + 2 * kh;
    v2f a0 = *(const v2f*)(Ar0 + ko);
    v2f a1 = *(const v2f*)(Ar1 + ko);
    v2f b0 = *(const v2f*)(Br0 + ko);
    v2f b1 = *(const v2f*)(Br1 + ko);
    v2f b2 = *(const v2f*)(Br2 + ko);
    v2f b3 = *(const v2f*)(Br3 + ko);
    acc[0][0] = __builtin_amdgcn_wmma_f32_16x16x4_f32(false, a0, false, b0, (short)0, acc[0][0], false, false);
    acc[0][1] = __builtin_amdgcn_wmma_f32_16x16x4_f32(false, a0, false, b1, (short)0, acc[0][1], false, false);
    acc[0][2] = __builtin_amdgcn_wmma_f32_16x16x4_f32(false, a0, false, b2, (short)0, acc[0][2], false, false);
    acc[0][3] = __builtin_amdgcn_wmma_f32_16x16x4_f32(false, a0, false, b3, (short)0, acc[0][3], false, false);
    acc[1][0] = __builtin_amdgcn_wmma_f32_16x16x4_f32(false, a1, false, b0, (short)0, acc[1][0], false, false);
    acc[1][1] = __builtin_amdgcn_wmma_f32_16x16x4_f32(false, a1, false, b1, (short)0, acc[1][1], false, false);
    acc[1][2] = __builtin_amdgcn_wmma_f32_16x16x4_f32(false, a1, false, b2, (short)0, acc[1][2], false, false);
    acc[1][3] = __builtin_amdgcn_wmma_f32_16x16x4_f32(false, a1, false, b3, (short)0, acc[1][3], false, false);
  }

  #pragma unroll
  for (int ti = 0; ti < 2; ++ti)
    #pragma unroll
    for (int tj = 0; tj < 4; ++tj) {
      const int m0 = mBase + ti * 16 + 8 * kh;  // lanes 16-31 hold M=v+8
      const int n0 = nBase + tj * 16 + r;
      #pragma unroll
      for (int v = 0; v < 8; ++v)
        C[(size_t)(m0 + v) * N + n0] = acc[ti][tj][v];
    }
}

// ---------------------------------------------------------------------------
// Depthwise causal conv (width 4) + bias + SiLU over the xc half of xz.
// ---------------------------------------------------------------------------
__global__ void conv_silu_kernel(const float* __restrict__ xz,
                                 const float* __restrict__ cw,
                                 const float* __restrict__ cb,
                                 float* __restrict__ xc,
                                 int Bn, int L, int Din, int ldxz) {
  size_t idx = (size_t)blockIdx.x * blockDim.x + threadIdx.x;
  size_t total = (size_t)Bn * L * Din;
  if (idx >= total) return;
  int d = (int)(idx % Din);
  int l = (int)((idx / Din) % L);
  int b = (int)(idx / ((size_t)Din * L));
  float sum = cb[d];
  #pragma unroll
  for (int j = 0; j < 4; ++j) {
    int ll = l - 3 + j;
    if (ll >= 0)
      sum += cw[d * 4 + j] * xz[((size_t)b * L + ll) * ldxz + d];
  }
  float sg = 1.0f / (1.0f + __expf(-sum));
  xc[idx] = sum * sg;
}

// ---------------------------------------------------------------------------
// x_proj: xp[row, o] = dot(xc_silu[row, :], x_proj_w[o, :]), o in [0,33).
// One wave32 per (row, o); shfl reduction. N=33 is too skinny for WMMA.
// ---------------------------------------------------------------------------
__global__ __launch_bounds__(256) void xproj_kernel(
    const float* __restrict__ xc, const float* __restrict__ w,
    float* __restrict__ xp, int rows, int Din) {
  int widx = blockIdx.x * 8 + (threadIdx.x >> 5);
  int lane = threadIdx.x & 31;
  int row = widx / 33;
  int o   = widx % 33;
  if (row >= rows) return;
  const float* xr = xc + (size_t)row * Din;
  const float* wr = w  + (size_t)o   * Din;
  float p = 0.f;
  for (int d = lane; d < Din; d += 32) p += xr[d] * wr[d];
  #pragma unroll
  for (int off = 16; off > 0; off >>= 1) p += __shfl_down(p, off);
  if (lane == 0) xp[(size_t)row * 33 + o] = p;
}

// ---------------------------------------------------------------------------
// Chunked parallel scan. L is split into NSEG segments of SL.
// h_t = a_t*h_{t-1} + u_t with diagonal a_t = exp(delta_t*A[d,s]),
// u_t = delta_t*B_t[s]*xc_t[d]. Three passes:
//   pass1: per (b,d,seg): P[s]=prod(a_t), Q[s]=local end state (h_start=0)
//   pass2: per (b,d,s):  sequential scan over the 16 segment summaries -> H0
//   pass3: per (b,d,seg): redo segment from true H0, emit y + fused epilogue
// Parallelism goes from 128 waves (monolithic scan) to 2048 waves.
// ---------------------------------------------------------------------------
#define SL    128
#define NSEG  16

__device__ __forceinline__ float softplus_f(float x) {
  return (x > 20.f) ? x : __logf(1.f + __expf(x));
}

__global__ __launch_bounds__(256) void scan_pass1(
    const float* __restrict__ xp, const float* __restrict__ xc,
    const float* __restrict__ A_log,
    float* __restrict__ P, float* __restrict__ Q, int L, int Din) {
  __shared__ float sxp[SL * 33];
  const int d   = blockIdx.x * blockDim.x + threadIdx.x;
  const int seg = blockIdx.y;
  const int b   = blockIdx.z;
  const int t0  = seg * SL;

  const float* src = xp + ((size_t)b * L + t0) * 33;
  for (int i = threadIdx.x; i < SL * 33; i += blockDim.x) sxp[i] = src[i];
  __syncthreads();

  float Areg[16];
  #pragma unroll
  for (int s = 0; s < 16; ++s) Areg[s] = -__expf(A_log[d * 16 + s]);

  float h[16], p[16];
  #pragma unroll
  for (int s = 0; s < 16; ++s) { h[s] = 0.f; p[s] = 1.f; }

  for (int tt = 0; tt < SL; ++tt) {
    const float* row = &sxp[tt * 33];                 // broadcast reads
    const float delta = softplus_f(row[0]);
    const float xct = xc[((size_t)b * L + t0 + tt) * Din + d];
    const float dx  = delta * xct;
    #pragma unroll
    for (int s = 0; s < 16; ++s) {
      float a = __expf(delta * Areg[s]);
      h[s] = a * h[s] + dx * row[1 + s];
      p[s] *= a;
    }
  }
  const size_t base = (((size_t)b * NSEG + seg) * Din + d) * 16;
  #pragma unroll
  for (int s = 0; s < 16; ++s) { P[base + s] = p[s]; Q[base + s] = h[s]; }
}

__global__ __launch_bounds__(256) void scan_pass2(
    const float* __restrict__ P, const float* __restrict__ Q,
    float* __restrict__ H0, int Din) {
  const int idx = blockIdx.x * blockDim.x + threadIdx.x;  // over B*Din*16
  const int b  = idx / (Din * 16);
  const int ds = idx % (Din * 16);
  float h = 0.f;
  #pragma unroll
  for (int seg = 0; seg < NSEG; ++seg) {
    const size_t o = ((size_t)b * NSEG + seg) * Din * 16 + ds;
    H0[o] = h;
    h = P[o] * h + Q[o];
  }
}

__global__ __launch_bounds__(256) void scan_pass3(
    const float* __restrict__ xp, const float* __restrict__ xc,
    const float* __restrict__ xz, const float* __restrict__ A_log,
    const float* __restrict__ Dp, const float* __restrict__ H0,
    float* __restrict__ yout, int L, int Din, int ldxz) {
  __shared__ float sxp[SL * 33];
  const int d   = blockIdx.x * blockDim.x + threadIdx.x;
  const int seg = blockIdx.y;
  const int b   = blockIdx.z;
  const int t0  = seg * SL;

  const float* src = xp + ((size_t)b * L + t0) * 33;
  for (int i = threadIdx.x; i < SL * 33; i += blockDim.x) sxp[i] = src[i];
  __syncthreads();

  float Areg[16];
  #pragma unroll
  for (int s = 0; s < 16; ++s) Areg[s] = -__expf(A_log[d * 16 + s]);
  const float Dreg = Dp[d];

  float h[16];
  const size_t base = (((size_t)b * NSEG + seg) * Din + d) * 16;
  #pragma unroll
  for (int s = 0; s < 16; ++s) h[s] = H0[base + s];

  for (int tt = 0; tt < SL; ++tt) {
    const int t = t0 + tt;
    const float* row = &sxp[tt * 33];
    const float delta = softplus_f(row[0]);
    const float xct = xc[((size_t)b * L + t) * Din + d];
    const float zt  = xz[((size_t)b * L + t) * ldxz + Din + d];
    const float dx  = delta * xct;
    float y = 0.f;
    #pragma unroll
    for (int s = 0; s < 16; ++s) {
      float a = __expf(delta * Areg[s]);
      h[s] = a * h[s] + dx * row[1 + s];
      y += h[s] * row[17 + s];
    }
    y += xct * Dreg;
    float sz = zt / (1.f + __expf(-zt));
    yout[((size_t)b * L + t) * Din + d] = y * sz;
  }
}

// ---------------------------------------------------------------------------
// Launcher. Workspace layout (floats):
//   xz      : rows*4096        (64 MB)
//   xc_silu : rows*2048        (32 MB)
//   xp      : rows*33          (0.5 MB)
//   y_final : rows*2048        (32 MB)
//   P,Q,H0  : 3 * B*NSEG*Din*16 (12 MB)
// ---------------------------------------------------------------------------
extern "C" void kernel_launch(void* const* d_in, const int* in_sizes, int n_in,
                              void* d_out, int out_size, void* d_ws, size_t ws_size,
                              hipStream_t stream) {
  const float* x          = (const float*)d_in[0];
  const float* in_proj_w  = (const float*)d_in[1];
  const float* conv_w     = (const float*)d_in[2];
  const float* conv_b     = (const float*)d_in[3];
  const float* x_proj_w   = (const float*)d_in[4];
  const float* A_log      = (const float*)d_in[5];
  const float* D_param    = (const float*)d_in[6];
  const float* out_proj_w = (const float*)d_in[7];
  float* out = (float*)d_out;

  const int Bn = 2, L = 2048, Dm = 1024, Din = 2048;
  const int rows = Bn * L;                 // 4096
  const int N1 = 2 * Din;                  // 4096 (xz width)
  const size_t segElems = (size_t)Bn * NSEG * Din * 16;

  float* ws  = (float*)d_ws;
  float* xz  = ws;
  float* xcs = xz  + (size_t)rows * N1;
  float* xpv = xcs + (size_t)rows * Din;
  float* yf  = xpv + (size_t)rows * 33;
  float* Pw  = yf  + (size_t)rows * Din;
  float* Qw  = Pw  + segElems;
  float* H0w = Qw  + segElems;

  // 1) xz = x @ in_proj_w.T : M=4096, N=4096, K=1024
  {
    dim3 g(rows / 128, N1 / 128);
    wmma_gemm_f32<<<g, 256, 0, stream>>>(x, in_proj_w, xz, rows, N1, Dm);
  }

  // 2) depthwise conv + SiLU on xc half of xz
  {
    size_t tot = (size_t)rows * Din;
    conv_silu_kernel<<<(unsigned)((tot + 255) / 256), 256, 0, stream>>>(
        xz, conv_w, conv_b, xcs, Bn, L, Din, N1);
  }

  // 3) xp = xc_silu @ x_proj_w.T (N=33)
  {
    int nwaves = rows * 33;
    xproj_kernel<<<(nwaves + 7) / 8, 256, 0, stream>>>(xcs, x_proj_w, xpv, rows, Din);
  }

  // 4) chunked parallel scan -> y_final
  {
    dim3 g13(Din / 256, NSEG, Bn);
    scan_pass1<<<g13, 256, 0, stream>>>(xpv, xcs, A_log, Pw, Qw, L, Din);
    int tot2 = Bn * Din * 16;
    scan_pass2<<<tot2 / 256, 256, 0, stream>>>(Pw, Qw, H0w, Din);
    scan_pass3<<<g13, 256, 0, stream>>>(xpv, xcs, xz, A_log, D_param, H0w, yf,
                                        L, Din, N1);
  }

  // 5) out = y_final @ out_proj_w.T : M=4096, N=1024, K=2048
  {
    dim3 g(rows / 128, Dm / 128);
    wmma_gemm_f32<<<g, 256, 0, stream>>>(yf, out_proj_w, out, rows, Dm, Din);
  }
}